// VocabGraphConvolution_77034533421631
// MI455X (gfx1250) — compile-verified
//
#include <hip/hip_runtime.h>

// Problem constants (match reference)
#define BATCH 32
#define V     1024
#define HID   768
#define OUT   256
#define S1    8     // row-split for adj column sums
#define S2    8     // w-split for weighted gather

typedef __attribute__((ext_vector_type(2))) float v2f;
typedef __attribute__((ext_vector_type(8))) float v8f;

// ---------------------------------------------------------------------------
// K1: partial column sums of adj.  colpart[b,s,w] = sum over 128 rows of adj[b].
// Fully coalesced 1 KB row-segment reads per block; pure HBM streaming.
// ---------------------------------------------------------------------------
__global__ void colsum_partial_kernel(const float* __restrict__ adj,
                                      float* __restrict__ part) {
  const int b = blockIdx.x;
  const int w = blockIdx.y * 256 + threadIdx.x;
  const int s = blockIdx.z;
  const float* p = adj + (size_t)b * V * V + (size_t)(s * (V / S1)) * V + w;
  float acc = 0.0f;
#pragma unroll 4
  for (int v = 0; v < V / S1; ++v) {
    __builtin_prefetch(p + 16 * V, 0, 1);   // global_prefetch_b8, ~64KB ahead
    acc += *p;
    p += V;
  }
  part[((b * S1 + s) * V) + w] = acc;
}

// K2: deterministic reduction of the S1 partials -> colsum[b,w]
__global__ void colsum_reduce_kernel(const float* __restrict__ part,
                                     float* __restrict__ colsum) {
  const int b = blockIdx.x;
  const int w = blockIdx.y * 256 + threadIdx.x;
  float acc = 0.0f;
#pragma unroll
  for (int s = 0; s < S1; ++s) acc += part[(b * S1 + s) * V + w];
  colsum[b * V + w] = acc;
}

// ---------------------------------------------------------------------------
// K3: hpart[b,s,h] = sum_{w in slice s} colsum[b,w] * emb[gv_ids[b,w], h]
// id/colsum are block-uniform (scalar loads); embedding row segments are
// coalesced 1 KB reads, mostly L2-resident (table 154 MB vs 192 MB L2).
// ---------------------------------------------------------------------------
__global__ void gather_partial_kernel(const float* __restrict__ emb,
                                      const int* __restrict__ ids,
                                      const float* __restrict__ colsum,
                                      float* __restrict__ hpart) {
  const int b  = blockIdx.x;
  const int hh = blockIdx.y * 256 + threadIdx.x;
  const int s  = blockIdx.z;
  const int wbase = s * (V / S2);
  float acc = 0.0f;
  for (int i = 0; i < V / S2; ++i) {
    const int   w  = wbase + i;
    const int   id = ids[b * V + w];          // uniform -> s_load
    const float cw = colsum[b * V + w];       // uniform -> s_load
    acc += cw * emb[(size_t)id * HID + hh];
  }
  hpart[(b * S2 + s) * HID + hh] = acc;
}

// K4: deterministic reduction + mean scale -> h[b,hid]
__global__ void h_reduce_kernel(const float* __restrict__ hpart,
                                float* __restrict__ hbuf) {
  const int b  = blockIdx.x;
  const int hh = blockIdx.y * 256 + threadIdx.x;
  float acc = 0.0f;
#pragma unroll
  for (int s = 0; s < S2; ++s) acc += hpart[(b * S2 + s) * HID + hh];
  hbuf[b * HID + hh] = acc * (1.0f / (float)V);
}

// ---------------------------------------------------------------------------
// K5: out[32,256] = h[32,768] @ fc_w[256,768]^T + fc_b  via V_WMMA_F32_16X16X4_F32
// One wave per 16x16 output tile: 2 M-tiles x 16 N-tiles = 32 waves = 4 blocks.
// A layout: lane l<16 -> A[l][k],A[l][k+1]; lane l>=16 -> A[l-16][k+2],A[l-16][k+3]
// B layout mirrors A (low lanes K=0,1; high lanes K=2,3; N striped over 16 lanes).
// Both A (h row-major) and B (fc_w row-major = B^T) load as [row][k + 2*half].
// ---------------------------------------------------------------------------
__global__ void fc_wmma_kernel(const float* __restrict__ hbuf,
                               const float* __restrict__ fc_w,
                               const float* __restrict__ fc_b,
                               float* __restrict__ out) {
  const int wave  = (blockIdx.x * blockDim.x + threadIdx.x) >> 5;
  const int lane  = threadIdx.x & 31;
  const int mTile = wave >> 4;     // 0..1
  const int nTile = wave & 15;     // 0..15
  const int m0 = mTile * 16;
  const int n0 = nTile * 16;
  const int half = lane >> 4;      // 0: K pair {0,1}, 1: K pair {2,3}
  const int l    = lane & 15;
  const int kOff = 2 * half;

  const float* arow = hbuf + (size_t)(m0 + l) * HID;   // A row m0+l
  const float* brow = fc_w + (size_t)(n0 + l) * HID;   // B col n0+l (fc_w row)

  v8f c = {};
  for (int k = 0; k < HID; k += 4) {
    v2f a  = *(const v2f*)(arow + k + kOff);   // 8-byte aligned: (k+2*half)*4 % 8 == 0
    v2f bb = *(const v2f*)(brow + k + kOff);
    // 8 args: (neg_a, A, neg_b, B, c_mod, C, reuse_a, reuse_b)
    c = __builtin_amdgcn_wmma_f32_16x16x4_f32(false, a, false, bb,
                                              (short)0, c, false, false);
  }

  // D layout: vgpr i, lanes 0-15 -> (M=i, N=l); lanes 16-31 -> (M=i+8, N=l)
#pragma unroll
  for (int i = 0; i < 8; ++i) {
    const int m = m0 + i + 8 * half;
    const int n = n0 + l;
    out[m * OUT + n] = c[i] + fc_b[n];
  }
}

extern "C" void kernel_launch(void* const* d_in, const int* in_sizes, int n_in,
                              void* d_out, int out_size, void* d_ws, size_t ws_size,
                              hipStream_t stream) {
  (void)in_sizes; (void)n_in; (void)out_size; (void)ws_size;
  const float* emb    = (const float*)d_in[0];  // [VOCAB, HID]
  const float* adj    = (const float*)d_in[1];  // [B, V, V]
  const float* fc_w   = (const float*)d_in[2];  // [OUT, HID]
  const float* fc_b   = (const float*)d_in[3];  // [OUT]
  const int*   gv_ids = (const int*)d_in[4];    // [B, V]
  float* out = (float*)d_out;                   // [B, OUT]

  // Workspace layout (floats): ~2 MB total
  char* ws = (char*)d_ws;
  float* part   = (float*)ws;                                               // B*S1*V
  float* colsum = (float*)(ws + sizeof(float) * ((size_t)BATCH * S1 * V));  // B*V
  float* hpart  = (float*)(ws + sizeof(float) * ((size_t)BATCH * S1 * V +
                                                 (size_t)BATCH * V));       // B*S2*HID
  float* hbuf   = (float*)(ws + sizeof(float) * ((size_t)BATCH * S1 * V +
                                                 (size_t)BATCH * V +
                                                 (size_t)BATCH * S2 * HID)); // B*HID

  colsum_partial_kernel<<<dim3(BATCH, V / 256, S1), 256, 0, stream>>>(adj, part);
  colsum_reduce_kernel <<<dim3(BATCH, V / 256),     256, 0, stream>>>(part, colsum);
  gather_partial_kernel<<<dim3(BATCH, HID / 256, S2), 256, 0, stream>>>(emb, gv_ids, colsum, hpart);
  h_reduce_kernel      <<<dim3(BATCH, HID / 256),     256, 0, stream>>>(hpart, hbuf);
  // 32 waves total = 4 blocks x 8 waves
  fc_wmma_kernel<<<dim3(4), 256, 0, stream>>>(hbuf, fc_w, fc_b, out);
}